// ViewDependentColorMLP_20091857011542
// MI455X (gfx1250) — compile-verified
//
#include <hip/hip_runtime.h>

typedef float v2f __attribute__((ext_vector_type(2)));
typedef float v8f __attribute__((ext_vector_type(8)));

#define F_DIM 32
#define E_DIM 16
#define NB    16
#define DIN   64      // E + F + NB
#define WID   128     // hidden width
#define XS    68      // xbuf row stride (floats): banks = 4*row + k  -> conflict-free b64 A reads
#define HS    132     // hbuf row stride (floats): banks = 4*row + k  -> conflict-free
#define WPS   288     // packed weight stride per k-pair: banks = 32*pk + 2*col -> half-waves disjoint
#define W2S   16

__global__ __launch_bounds__(256, 1)
void vdc_mlp_wmma(const float* __restrict__ features,
                  const float* __restrict__ dirs,
                  const float* __restrict__ embed_table,
                  const float* __restrict__ W0g,
                  const float* __restrict__ b0g,
                  const float* __restrict__ W1g,
                  const float* __restrict__ b1g,
                  const float* __restrict__ W2g,
                  const float* __restrict__ b2g,
                  const int*   __restrict__ embed_ids,
                  const int*   __restrict__ sh_degree,
                  float*       __restrict__ out,
                  int N, int CN)
{
    __shared__ float xs[256 * XS];        // 69,632 B   input rows
    __shared__ float hb[8 * 16 * HS];     // 67,584 B   per-wave 16-row activation patch
    __shared__ float lw0[32 * WPS];       // 36,864 B   W0 packed (k-pairs)
    __shared__ float lw1[64 * WPS];       // 73,728 B   W1 packed
    __shared__ float lw2[WID * W2S];      //  8,192 B   W2 padded 128x16
    __shared__ float lb0[WID], lb1[WID], lb2[16];

    const int tid = threadIdx.x;

    // ---------------- cooperative weight load (B-operand-ready packing) ----------------
    for (int i = tid; i < DIN * WID; i += 256) {
        int k = i / WID, n = i % WID;
        lw0[(k >> 1) * WPS + n * 2 + (k & 1)] = W0g[i];
    }
    for (int i = tid; i < WID * WID; i += 256) {
        int k = i / WID, n = i % WID;
        lw1[(k >> 1) * WPS + n * 2 + (k & 1)] = W1g[i];
    }
    for (int i = tid; i < WID * W2S; i += 256) {
        int k = i / W2S, c = i % W2S;
        lw2[i] = (c < 3) ? W2g[k * 3 + c] : 0.0f;
    }
    if (tid < WID) { lb0[tid] = b0g[tid]; lb1[tid] = b1g[tid]; }
    if (tid < 16)  { lb2[tid] = (tid < 3) ? b2g[tid] : 0.0f; }

    // ---------------- build one input row per thread: [emb | feats | SH] ----------------
    const int g = blockIdx.x * 256 + tid;
    {
        float xr[DIN];
        if (g < CN) {
            const int cam = g / N;
            const int n   = g - cam * N;
            const int eid = embed_ids[cam];
            #pragma unroll
            for (int j = 0; j < E_DIM; ++j) xr[j] = embed_table[eid * E_DIM + j];
            #pragma unroll
            for (int j = 0; j < F_DIM; ++j) xr[E_DIM + j] = features[n * F_DIM + j];

            float dx = dirs[3 * g + 0], dy = dirs[3 * g + 1], dz = dirs[3 * g + 2];
            float nrm = sqrtf(dx * dx + dy * dy + dz * dz);
            float inv = 1.0f / fmaxf(nrm, 1e-12f);
            float X = dx * inv, Y = dy * inv, Z = dz * inv;
            float z2 = Z * Z;
            float fTmp0B = -1.092548430592079f * Z;
            float fC1 = X * X - Y * Y;
            float fS1 = 2.0f * X * Y;
            float fTmp0C = -2.285228997322329f * z2 + 0.4570457994644658f;
            float fTmp1B = 1.445305721320277f * Z;
            float fC2 = X * fC1 - Y * fS1;
            float fS2 = X * fS1 + Y * fC1;
            float* sh = xr + (DIN - NB);
            sh[0]  = 0.2820947917738781f;
            sh[1]  = -0.48860251190292f * Y;
            sh[2]  =  0.48860251190292f * Z;
            sh[3]  = -0.48860251190292f * X;
            sh[4]  = 0.5462742152960395f * fS1;
            sh[5]  = fTmp0B * Y;
            sh[6]  = 0.9461746957575601f * z2 - 0.3153915652525201f;
            sh[7]  = fTmp0B * X;
            sh[8]  = 0.5462742152960395f * fC1;
            sh[9]  = -0.5900435899266435f * fS2;
            sh[10] = fTmp1B * fS1;
            sh[11] = fTmp0C * Y;
            sh[12] = Z * (1.865881662950577f * z2 - 1.119528997770346f);
            sh[13] = fTmp0C * X;
            sh[14] = fTmp1B * fC1;
            sh[15] = -0.5900435899266435f * fC2;
            int nb = *sh_degree + 1; nb = nb * nb;     // zero bases beyond requested degree
            #pragma unroll
            for (int j = 0; j < NB; ++j) if (j >= nb) sh[j] = 0.0f;
        } else {
            #pragma unroll
            for (int j = 0; j < DIN; ++j) xr[j] = 0.0f;
        }
        #pragma unroll
        for (int j = 0; j < DIN; ++j) xs[tid * XS + j] = xr[j];
    }
    __syncthreads();   // weights visible to everyone (x rows are wave-local anyway)

    // ---------------- WMMA MLP: each wave owns two 16-row tiles ----------------
    const int wave = tid >> 5;
    const int lane = tid & 31;
    const int m    = lane & 15;     // C/D: column | A: row
    const int hi   = lane >> 4;     // half-wave selector (k-offset 2*hi)
    const int hrb  = wave * 16;     // wave-private hbuf row base

    for (int tile = 0; tile < 2; ++tile) {
        const int rb = wave * 32 + tile * 16;          // block-local row base

        // ----- layer 0: x[64] @ W0[64x128] + b0, ReLU -----
        v8f acc[8];
        #pragma unroll
        for (int nt = 0; nt < 8; ++nt) {
            float bv = lb0[nt * 16 + m];
            #pragma unroll
            for (int r = 0; r < 8; ++r) acc[nt][r] = bv;
        }
        {
            const float* arow = &xs[(rb + m) * XS + 2 * hi];
            for (int kk = 0; kk < DIN; kk += 4) {
                v2f a = *(const v2f*)(arow + kk);
                const float* bp = &lw0[((kk >> 1) + hi) * WPS + m * 2];
                #pragma unroll
                for (int nt = 0; nt < 8; ++nt) {
                    v2f b = *(const v2f*)(bp + nt * 32);
                    acc[nt] = __builtin_amdgcn_wmma_f32_16x16x4_f32(
                        false, a, false, b, (short)0, acc[nt], false, false);
                }
            }
        }
        #pragma unroll
        for (int nt = 0; nt < 8; ++nt)
            #pragma unroll
            for (int r = 0; r < 8; ++r)
                hb[(hrb + r + 8 * hi) * HS + nt * 16 + m] = fmaxf(acc[nt][r], 0.0f);

        // ----- layer 1: h0[128] @ W1[128x128] + b1, ReLU -----
        v8f acc2[8];
        #pragma unroll
        for (int nt = 0; nt < 8; ++nt) {
            float bv = lb1[nt * 16 + m];
            #pragma unroll
            for (int r = 0; r < 8; ++r) acc2[nt][r] = bv;
        }
        const float* hrow = &hb[(hrb + m) * HS + 2 * hi];
        for (int kk = 0; kk < WID; kk += 4) {
            v2f a = *(const v2f*)(hrow + kk);
            const float* bp = &lw1[((kk >> 1) + hi) * WPS + m * 2];
            #pragma unroll
            for (int nt = 0; nt < 8; ++nt) {
                v2f b = *(const v2f*)(bp + nt * 32);
                acc2[nt] = __builtin_amdgcn_wmma_f32_16x16x4_f32(
                    false, a, false, b, (short)0, acc2[nt], false, false);
            }
        }
        #pragma unroll
        for (int nt = 0; nt < 8; ++nt)
            #pragma unroll
            for (int r = 0; r < 8; ++r)
                hb[(hrb + r + 8 * hi) * HS + nt * 16 + m] = fmaxf(acc2[nt][r], 0.0f);

        // ----- layer 2: h1[128] @ W2[128x3] + b2 (W2 padded to 16 cols) -----
        v8f acc3;
        {
            float bv = lb2[m];
            #pragma unroll
            for (int r = 0; r < 8; ++r) acc3[r] = bv;
        }
        for (int kk = 0; kk < WID; kk += 4) {
            v2f a = *(const v2f*)(hrow + kk);
            v2f b;
            b.x = lw2[(kk + 2 * hi) * W2S + m];
            b.y = lw2[(kk + 2 * hi + 1) * W2S + m];
            acc3 = __builtin_amdgcn_wmma_f32_16x16x4_f32(
                false, a, false, b, (short)0, acc3, false, false);
        }

        // ----- store: lane (m<3) holds out channel m of splat (r + 8*hi) -----
        if (m < 3) {
            const int base = blockIdx.x * 256 + rb;
            #pragma unroll
            for (int r = 0; r < 8; ++r) {
                int row = base + r + 8 * hi;
                if (row < CN) out[row * 3 + m] = acc3[r];
            }
        }
    }
}

extern "C" void kernel_launch(void* const* d_in, const int* in_sizes, int n_in,
                              void* d_out, int out_size, void* d_ws, size_t ws_size,
                              hipStream_t stream) {
    const float* features    = (const float*)d_in[0];
    const float* dirs        = (const float*)d_in[1];
    const float* embed_table = (const float*)d_in[2];
    const float* W0g         = (const float*)d_in[3];
    const float* b0g         = (const float*)d_in[4];
    const float* W1g         = (const float*)d_in[5];
    const float* b1g         = (const float*)d_in[6];
    const float* W2g         = (const float*)d_in[7];
    const float* b2g         = (const float*)d_in[8];
    const int*   embed_ids   = (const int*)d_in[9];
    const int*   sh_degree   = (const int*)d_in[10];
    float*       out         = (float*)d_out;

    const int C  = in_sizes[9];
    const int N  = in_sizes[0] / F_DIM;
    const int CN = C * N;
    const int grid = (CN + 255) / 256;

    hipLaunchKernelGGL(vdc_mlp_wmma, dim3(grid), dim3(256), 0, stream,
                       features, dirs, embed_table, W0g, b0g, W1g, b1g,
                       W2g, b2g, embed_ids, sh_degree, out, N, CN);
}